// Softmax_19473381720488
// MI455X (gfx1250) — compile-verified
//
#include <hip/hip_runtime.h>
#include <stdint.h>

// Segment-wise numerically-stable softmax (torch_geometric.utils.softmax style).
// x: [N, 128] f32, batch: [N] sorted segment ids in [0, 2048), out: [N, 128] f32.

typedef float        f32x4 __attribute__((ext_vector_type(4)));
typedef unsigned int u32x4 __attribute__((ext_vector_type(4)));
typedef int          i32x4 __attribute__((ext_vector_type(4)));
typedef int          i32x8 __attribute__((ext_vector_type(8)));

#define NSEG   2048
#define HDIM   128
#define WAVES  8
#define BLOCK  (WAVES * 32)
#define TILE   48              // rows per TDM tile (48 * 512B = 24 KB per buffer)
#define LOG2E  1.4426950408889634f
#define EPSV   1e-16f

__device__ __forceinline__ f32x4 max4(f32x4 a, f32x4 b) {
    f32x4 r;
    r.x = __builtin_fmaxf(a.x, b.x);
    r.y = __builtin_fmaxf(a.y, b.y);
    r.z = __builtin_fmaxf(a.z, b.z);
    r.w = __builtin_fmaxf(a.w, b.w);
    return r;
}

__device__ __forceinline__ f32x4 exp2_4(f32x4 a) {
    f32x4 r;
    r.x = __builtin_amdgcn_exp2f(a.x);
    r.y = __builtin_amdgcn_exp2f(a.y);
    r.z = __builtin_amdgcn_exp2f(a.z);
    r.w = __builtin_amdgcn_exp2f(a.w);
    return r;
}

__device__ __forceinline__ int lower_bound(const int* __restrict__ b, int n, int key) {
    int lo = 0, hi = n;
    while (lo < hi) {
        int mid = (lo + hi) >> 1;
        if (b[mid] < key) lo = mid + 1; else hi = mid;
    }
    return lo;
}

// Issue a TDM load of `rows` consecutive 128-float rows starting at gsrc into LDS
// at byte offset ldsByteAddr. 2-D tile: tile_dim0 = 128 elems (one row, 512 B),
// tile_dim1 = rows, row stride = 128 elems. D# layout per CDNA5 ISA ch.8.
__device__ __forceinline__ void tdm_load_rows(const float* gsrc,
                                              unsigned ldsByteAddr, int rows) {
    const unsigned long long ga = (unsigned long long)(uintptr_t)gsrc;
    u32x4 g0;
    g0.x = 1u;                                   // count=1 (valid), user mode
    g0.y = ldsByteAddr;                          // lds_addr [63:32]
    g0.z = (unsigned)(ga & 0xffffffffull);       // global_addr [95:64]
    g0.w = ((unsigned)(ga >> 32) & 0x01ffffffu)  // global_addr [120:96]
         | 0x80000000u;                          // type=2 ("image") [127:126]

    i32x8 g1;
    g1[0] = 0x00020000;                          // wg_mask=0, data_size=2 (4 B)
    g1[1] = (int)(128u << 16);                   // tensor_dim0 = 128 (lo16 in [63:48])
    g1[2] = (int)((unsigned)rows << 16);         // tensor_dim0 hi=0; tensor_dim1 lo16
    g1[3] = (int)(128u << 16);                   // tensor_dim1 hi=0; tile_dim0 = 128
    g1[4] = rows;                                // tile_dim1 = rows; tile_dim2 = 0
    g1[5] = 128;                                 // tensor_dim0_stride = 128 elems
    g1[6] = 0;                                   // stride hi; tensor_dim1_stride lo
    g1[7] = 0;                                   // tensor_dim1_stride hi

    const i32x4 z4 = (i32x4)0;                   // groups 2/3: <=2-D tensor, disabled
    const i32x8 z8 = (i32x8)0;                   // extra group (clang-23 6-arg form)
    __builtin_amdgcn_tensor_load_to_lds(g0, g1, z4, z4, z8, 0);
}

__global__ __launch_bounds__(BLOCK) void segsoftmax_kernel(
    const float* __restrict__ x,
    const int*   __restrict__ batch,
    float*       __restrict__ out,
    int n)
{
    __shared__ f32x4 tile[2][TILE][32];   // 2 x 24 KB TDM staging buffers
    __shared__ f32x4 red[BLOCK];          // 4 KB cross-wave reduction buffer

    const int s    = blockIdx.x;
    const int tid  = threadIdx.x;
    const int lane = tid & 31;
    const int wave = tid >> 5;

    // batch is sorted: segment s occupies contiguous rows [r0, r1).
    const int r0 = lower_bound(batch, n, s);
    const int r1 = lower_bound(batch, n, s + 1);
    if (r0 >= r1) return;   // empty segment: nothing to write (uniform exit)

    const size_t colOff = (size_t)(lane << 2);           // this lane's 4 columns
    const float* __restrict__ xs = x + colOff;
    float*       __restrict__ os = out + colOff;

    // LDS byte offsets of the two staging buffers (generic addr low 32 == LDS offset).
    const unsigned ldsBuf0 = (unsigned)(uintptr_t)(&tile[0][0][0]);
    const unsigned ldsBuf1 = (unsigned)(uintptr_t)(&tile[1][0][0]);

    // ---------------- Phase 1: per-column max (TDM double-buffered stream) ----------------
    f32x4 m = (f32x4)(-__builtin_inff());

    const int total = r1 - r0;
    const int nt    = (total + TILE - 1) / TILE;
    int p = 0;
    for (int t = 0; t < nt; ++t) {
        const int cur = r0 + t * TILE;
        const int cnt = (r1 - cur < TILE) ? (r1 - cur) : TILE;
        if (wave == 0) {
            if (t == 0)
                tdm_load_rows(x + (size_t)cur * HDIM, ldsBuf0, cnt);
            const int nxt = cur + TILE;
            if (nxt < r1) {
                const int ncnt = (r1 - nxt < TILE) ? (r1 - nxt) : TILE;
                tdm_load_rows(x + (size_t)nxt * HDIM, p ? ldsBuf0 : ldsBuf1, ncnt);
                __builtin_amdgcn_s_wait_tensorcnt(1);  // tensor ops complete in order
            } else {
                __builtin_amdgcn_s_wait_tensorcnt(0);
            }
        }
        __syncthreads();                       // tile t visible to all waves
        for (int i = wave; i < cnt; i += WAVES)
            m = max4(m, tile[p][i][lane]);     // ds_load_b128
        __syncthreads();                       // done reading before DMA reuses buffer
        p ^= 1;
    }

    // Cross-wave max reduction through LDS.
    red[tid] = m;
    __syncthreads();
    if (tid < 32) {
        f32x4 a = red[lane];
        #pragma unroll
        for (int w = 1; w < WAVES; ++w) a = max4(a, red[w * 32 + lane]);
        red[lane] = a;
    }
    __syncthreads();
    const f32x4 mx = red[lane];
    __syncthreads();

    // ---------------- Phase 2: per-column sum of exp (reads hit L2) ----------------
    f32x4 acc = (f32x4)(0.0f);
    #pragma unroll 4
    for (int r = r0 + wave; r < r1; r += WAVES) {
        f32x4 v = *(const f32x4*)(xs + (size_t)r * HDIM);
        acc += exp2_4((v - mx) * LOG2E);                       // v_exp_f32
    }
    red[tid] = acc;
    __syncthreads();
    if (tid < 32) {
        f32x4 a = red[lane];
        #pragma unroll
        for (int w = 1; w < WAVES; ++w) a += red[w * 32 + lane];
        red[lane] = a;
    }
    __syncthreads();
    const f32x4 ssum = red[lane];

    f32x4 inv;
    inv.x = 1.0f / (ssum.x + EPSV);
    inv.y = 1.0f / (ssum.y + EPSV);
    inv.z = 1.0f / (ssum.z + EPSV);
    inv.w = 1.0f / (ssum.w + EPSV);

    // ---------------- Phase 3: normalize + non-temporal store ----------------
    #pragma unroll 4
    for (int r = r0 + wave; r < r1; r += WAVES) {
        f32x4 v = *(const f32x4*)(xs + (size_t)r * HDIM);
        f32x4 o = exp2_4((v - mx) * LOG2E) * inv;
        __builtin_nontemporal_store(o, (f32x4*)(os + (size_t)r * HDIM));
    }
}

extern "C" void kernel_launch(void* const* d_in, const int* in_sizes, int n_in,
                              void* d_out, int out_size, void* d_ws, size_t ws_size,
                              hipStream_t stream) {
    const float* x     = (const float*)d_in[0];
    const int*   batch = (const int*)d_in[1];
    float*       out   = (float*)d_out;
    const int n = in_sizes[1];   // number of rows N (x has N*128 elements)

    segsoftmax_kernel<<<NSEG, BLOCK, 0, stream>>>(x, batch, out, n);
}